// GCN_88124138979416
// MI455X (gfx1250) — compile-verified
//
#include <hip/hip_runtime.h>
#include <hip/hip_bf16.h>

// ---------------------------------------------------------------------------
// GCN forward for MI455X (gfx1250, wave32).
//   layer1: m = segsum(feat[src]*dout^-1/2) ; h1 = relu((m*din^-1/2) @ W1 + b1)
//   layer2 (algebraically reordered, W2 is 512x1):
//       y[n]   = h1[n] . W2
//       out    = (1/N) * sum_e din[dst[e]]^-1/2 * dout[src[e]]^-1/2 * y[src[e]] + b2
// ---------------------------------------------------------------------------

#define N_NODES 50000
#define N_EDGES 400000
#define FEATS   512

typedef __attribute__((ext_vector_type(2))) float v2f;
typedef __attribute__((ext_vector_type(8))) float v8f;

// Workspace layout (float offsets). All regions zeroed each launch.
#define WS_DINV_OUT 0
#define WS_DINV_IN  50048
#define WS_S        100096
#define WS_ACC      150144
#define WS_M        150160                      // 16B aligned (150160*4 % 16 == 0)
#define WS_M_ELEMS  (N_NODES * FEATS)           // 25,600,000 floats
#define WS_ZERO_TOT (WS_M + WS_M_ELEMS)         // divisible by 4

// ---------------------------------------------------------------------------
__global__ void zero_kernel(float* ws, int n4) {
    int i = blockIdx.x * blockDim.x + threadIdx.x;
    if (i < n4) {
        ((float4*)ws)[i] = make_float4(0.f, 0.f, 0.f, 0.f);
    }
}

__global__ void degree_kernel(const int* __restrict__ src, const int* __restrict__ dst,
                              float* deg_out, float* deg_in) {
    int e = blockIdx.x * blockDim.x + threadIdx.x;
    if (e < N_EDGES) {
        atomicAdd(&deg_out[src[e]], 1.0f);
        atomicAdd(&deg_in[dst[e]], 1.0f);
    }
}

__global__ void rsqrt_kernel(float* deg_out, float* deg_in) {
    int n = blockIdx.x * blockDim.x + threadIdx.x;
    if (n < N_NODES) {
        deg_out[n] = rsqrtf(fmaxf(deg_out[n], 1.0f));
        deg_in[n]  = rsqrtf(fmaxf(deg_in[n], 1.0f));
    }
}

// One wave (32 lanes) per edge: m[dst] += feat[src] * dinv_out[src]
__global__ void spmm_kernel(const float* __restrict__ feat,
                            const int* __restrict__ src, const int* __restrict__ dst,
                            const float* __restrict__ dinv_out, float* __restrict__ m) {
    int tid  = blockIdx.x * blockDim.x + threadIdx.x;
    int e    = tid >> 5;
    int lane = tid & 31;
    if (e >= N_EDGES) return;
    int s = src[e];
    int d = dst[e];
    float scale = dinv_out[s];
    const float4* f4 = (const float4*)(feat + (size_t)s * FEATS);
    float* mrow = m + (size_t)d * FEATS;
#pragma unroll
    for (int j = 0; j < 4; ++j) {
        int i4 = lane + 32 * j;       // 128 float4 = 512 floats
        float4 v = f4[i4];
        atomicAdd(&mrow[i4 * 4 + 0], v.x * scale);
        atomicAdd(&mrow[i4 * 4 + 1], v.y * scale);
        atomicAdd(&mrow[i4 * 4 + 2], v.z * scale);
        atomicAdd(&mrow[i4 * 4 + 3], v.w * scale);
    }
}

// ---------------------------------------------------------------------------
// Tiled fp32 WMMA GEMM with fused scaling, bias and ReLU.
//   C[64x64 per block] = relu( (m * dinv_in[row]) @ W1 + b1 ), written in place
//   over the m buffer (each block reads only its own 64 rows before writing).
// Block = 128 threads = 4 waves; wave w owns rows [16w,16w+16) of the tile
// and all 4 16-wide column subtiles (4 x v8f accumulators).
// K loop staged through LDS in chunks of 32, consumed by V_WMMA_F32_16X16X4_F32.
// B is staged in fragment-ready layout Bs2[k/2][col][k&1] so each lane's
// (K, K+1) pair is one aligned 8-byte LDS load straight into the B fragment.
// ---------------------------------------------------------------------------
#define KT 32
__global__ __launch_bounds__(128)
void gemm_relu_kernel(float* __restrict__ A,            // m buffer [N_NODES, 512] (in/out)
                      const float* __restrict__ W1,     // [512, 512]
                      const float* __restrict__ b1,     // [512]
                      const float* __restrict__ dinv_in) {
    __shared__ __align__(16) float As[64][36];          // 144B row stride (16B multiple)
    __shared__ __align__(16) float Bs2[KT / 2][66][2];  // pair-major, 528B row stride

    const int m0 = blockIdx.x * 64;
    const int n0 = blockIdx.y * 64;
    const int t    = threadIdx.x;
    const int wave = t >> 5;
    const int lane = t & 31;
    const int hs   = lane >> 4;     // half-select: lanes 16-31 hold K+2 / M+8
    const int l15  = lane & 15;

    v8f acc[4];
#pragma unroll
    for (int j = 0; j < 4; ++j) acc[j] = (v8f)0.0f;

    for (int kk = 0; kk < FEATS; kk += KT) {
        // --- stage A tile: 64 rows x 32 cols, dinv_in folded in -------------
#pragma unroll
        for (int i = 0; i < 4; ++i) {
            int f   = t + 128 * i;          // float4 index in [0,512)
            int row = f >> 3;               // 8 float4 per row
            int c4  = f & 7;
            int rg  = m0 + row;
            float4 v = make_float4(0.f, 0.f, 0.f, 0.f);
            if (rg < N_NODES) {
                v = *(const float4*)(A + (size_t)rg * FEATS + kk + c4 * 4);
                float sc = dinv_in[rg];
                v.x *= sc; v.y *= sc; v.z *= sc; v.w *= sc;
            }
            *(float4*)&As[row][c4 * 4] = v;
        }
        // --- stage B tile: 32 rows x 64 cols of W1, pair-major --------------
#pragma unroll
        for (int i = 0; i < 4; ++i) {
            int f   = t + 128 * i;          // float4 index in [0,512)
            int row = f >> 4;               // k index in [0,32); 16 float4 per row
            int c4  = f & 15;
            float4 v = *(const float4*)(W1 + (size_t)(kk + row) * FEATS + n0 + c4 * 4);
            int kp = row >> 1, kl = row & 1;
            Bs2[kp][c4 * 4 + 0][kl] = v.x;
            Bs2[kp][c4 * 4 + 1][kl] = v.y;
            Bs2[kp][c4 * 4 + 2][kl] = v.z;
            Bs2[kp][c4 * 4 + 3][kl] = v.w;
        }
        __syncthreads();

        // hint-prefetch next A chunk for this wave's rows
        if (kk + KT < FEATS) {
            int rg = m0 + 16 * wave + l15;
            if (rg < N_NODES)
                __builtin_prefetch(A + (size_t)rg * FEATS + kk + KT, 0, 1);
        }

        // --- 8 WMMA K-steps of 4 -------------------------------------------
        const int rA = 16 * wave + l15;
#pragma unroll
        for (int k2 = 0; k2 < KT; k2 += 4) {
            // A frag (16x4 f32): lanes 0-15 -> K=k2+0,1 ; lanes 16-31 -> K=k2+2,3
            v2f a;
            a.x = As[rA][k2 + 2 * hs + 0];
            a.y = As[rA][k2 + 2 * hs + 1];
            const int kp = (k2 >> 1) + hs;  // pair row holding (K, K+1)
#pragma unroll
            for (int j = 0; j < 4; ++j) {
                v2f b = *(const v2f*)&Bs2[kp][16 * j + l15][0];  // one b64 load
                acc[j] = __builtin_amdgcn_wmma_f32_16x16x4_f32(
                    /*neg_a=*/false, a, /*neg_b=*/false, b,
                    /*c_mod=*/(short)0, acc[j],
                    /*reuse_a=*/false, /*reuse_b=*/false);
            }
        }
        __syncthreads();
    }

    // --- epilogue: bias + ReLU, store in place -----------------------------
#pragma unroll
    for (int j = 0; j < 4; ++j) {
        int col = n0 + 16 * j + l15;
        float bias = b1[col];
#pragma unroll
        for (int v = 0; v < 8; ++v) {
            int row = m0 + 16 * wave + v + 8 * hs;   // C layout: VGPR v -> M=v (+8 upper half)
            if (row < N_NODES) {
                float val = acc[j][v] + bias;
                A[(size_t)row * FEATS + col] = fmaxf(val, 0.0f);
            }
        }
    }
}

// One wave per node: s[n] = dinv_out[n] * dot(h1[n,:], W2[:,0])
__global__ void dot_kernel(const float* __restrict__ h1, const float* __restrict__ W2,
                           const float* __restrict__ dinv_out, float* __restrict__ s) {
    int tid  = blockIdx.x * blockDim.x + threadIdx.x;
    int n    = tid >> 5;
    int lane = tid & 31;
    if (n >= N_NODES) return;
    const float* row = h1 + (size_t)n * FEATS;
    float sum = 0.0f;
#pragma unroll
    for (int j = 0; j < 16; ++j) {
        int i = lane + 32 * j;
        sum += row[i] * W2[i];
    }
#pragma unroll
    for (int off = 16; off >= 1; off >>= 1)
        sum += __shfl_xor(sum, off, 32);
    if (lane == 0) s[n] = sum * dinv_out[n];
}

// acc += sum_e dinv_in[dst[e]] * s[src[e]]
__global__ void edge_reduce_kernel(const int* __restrict__ src, const int* __restrict__ dst,
                                   const float* __restrict__ s, const float* __restrict__ dinv_in,
                                   float* acc) {
    __shared__ float red[256];
    float local = 0.0f;
    for (int e = blockIdx.x * blockDim.x + threadIdx.x; e < N_EDGES;
         e += gridDim.x * blockDim.x) {
        local += dinv_in[dst[e]] * s[src[e]];
    }
    red[threadIdx.x] = local;
    __syncthreads();
    for (int st = 128; st >= 1; st >>= 1) {
        if (threadIdx.x < st) red[threadIdx.x] += red[threadIdx.x + st];
        __syncthreads();
    }
    if (threadIdx.x == 0) atomicAdd(acc, red[0]);
}

__global__ void final_kernel(const float* acc, const float* __restrict__ b2, float* out) {
    if (threadIdx.x == 0 && blockIdx.x == 0)
        out[0] = acc[0] * (1.0f / (float)N_NODES) + b2[0];
}

// ---------------------------------------------------------------------------
extern "C" void kernel_launch(void* const* d_in, const int* in_sizes, int n_in,
                              void* d_out, int out_size, void* d_ws, size_t ws_size,
                              hipStream_t stream) {
    const float* feat = (const float*)d_in[0];
    const int*   src  = (const int*)d_in[1];
    const int*   dst  = (const int*)d_in[2];
    const float* W1   = (const float*)d_in[3];
    const float* b1   = (const float*)d_in[4];
    const float* W2   = (const float*)d_in[5];
    const float* b2   = (const float*)d_in[6];
    float* out = (float*)d_out;

    float* wsf      = (float*)d_ws;
    float* dinv_out = wsf + WS_DINV_OUT;
    float* dinv_in  = wsf + WS_DINV_IN;
    float* s        = wsf + WS_S;
    float* acc      = wsf + WS_ACC;
    float* m        = wsf + WS_M;   // doubles as h1 after the GEMM

    // 1) zero degrees + accumulator + m buffer
    {
        int n4 = WS_ZERO_TOT / 4;
        zero_kernel<<<(n4 + 255) / 256, 256, 0, stream>>>(wsf, n4);
    }
    // 2) degrees
    degree_kernel<<<(N_EDGES + 255) / 256, 256, 0, stream>>>(src, dst, dinv_out, dinv_in);
    // 3) clamp + rsqrt (in place)
    rsqrt_kernel<<<(N_NODES + 255) / 256, 256, 0, stream>>>(dinv_out, dinv_in);
    // 4) SpMM aggregation (one wave per edge)
    spmm_kernel<<<(N_EDGES * 32 + 255) / 256, 256, 0, stream>>>(feat, src, dst, dinv_out, m);
    // 5) WMMA GEMM + bias + ReLU (in place over m)
    {
        dim3 grid((N_NODES + 63) / 64, FEATS / 64);
        gemm_relu_kernel<<<grid, 128, 0, stream>>>(m, W1, b1, dinv_in);
    }
    // 6) per-node dot with W2 column, scaled by dinv_out
    dot_kernel<<<(N_NODES * 32 + 255) / 256, 256, 0, stream>>>(m, W2, dinv_out, s);
    // 7) edge-level scalar reduction
    edge_reduce_kernel<<<512, 256, 0, stream>>>(src, dst, s, dinv_in, acc);
    // 8) mean + bias
    final_kernel<<<1, 32, 0, stream>>>(acc, b2, out);
}